// TimeEmbedder1_37022618092062
// MI455X (gfx1250) — compile-verified
//
#include <hip/hip_runtime.h>
#include <stddef.h>
#include <stdint.h>

// ---------------------------------------------------------------------------
// TimeEmbedder: out[i,:] = W[:, t_i] + b   (row gather, NOT a real GEMM)
//
// MI455X roofline: 1 GiB of output stores @ 23.3 TB/s => ~46us floor; W is
// 4 MB (L2-resident, 192 MB L2), so this is pure data movement. We use the
// CDNA5 async Global<->LDS DMA path (global_load_async_to_lds_b128 /
// global_store_async_from_lds_b128, ASYNCcnt) as the streaming engine after a
// one-time 4 MB transpose that folds the bias:  WtB[t][e] = W[e][t] + b[e].
//
// Pipeline: double-buffered LDS groups. Per iteration (one s_wait_asynccnt 0):
//   wait0  -> this group's loads landed AND previous group's stores (the last
//             readers of the other buffer) completed
//   issue stores(group i, buf)        ; HBM write stream drains async
//   issue loads (group i+1, buf^1)    ; L2 reads overlap the store drain
// ---------------------------------------------------------------------------

// ---- Prep kernel: LDS-tiled transpose of W [EMBED, MAXT] -> WtB [MAXT, EMBED]
// with bias folded in. Tiny (8 MB traffic total).
__global__ void te_transpose_bias(const float* __restrict__ W,
                                  const float* __restrict__ b,
                                  float* __restrict__ WtB,
                                  int maxt, int embed) {
  __shared__ float tile[32][33];  // +1 pad: no bank conflicts
  const int t0 = blockIdx.x * 32;
  const int e0 = blockIdx.y * 32;
  const int tx = threadIdx.x;     // 0..31
  const int ty = threadIdx.y;     // 0..7

  if (t0 + tx < maxt) {
#pragma unroll
    for (int k = 0; k < 4; ++k) {
      const int e = e0 + ty + k * 8;
      if (e < embed)
        tile[ty + k * 8][tx] = W[(size_t)e * maxt + (t0 + tx)];
    }
  }
  __syncthreads();
  if (e0 + tx < embed) {
    const float bias = b[e0 + tx];
#pragma unroll
    for (int k = 0; k < 4; ++k) {
      const int t = t0 + ty + k * 8;
      if (t < maxt)
        WtB[(size_t)t * embed + (e0 + tx)] = tile[tx][ty + k * 8] + bias;
    }
  }
}

// ---- Main kernel: out[r,:] = WtB[ts[r],:], rows of 4096 bytes.
// 256 threads/block; thread tid owns bytes [tid*16, tid*16+16) of each row.
// TE_RPB rows per group, two LDS buffers (2*6*4096 = 48 KB / block).
#define TE_RPB 6

__global__ void te_gather_rows_async(const float* __restrict__ WtB,
                                     const int* __restrict__ ts,
                                     float* __restrict__ out,
                                     int nrows) {
  __shared__ __align__(16) char smem[2 * TE_RPB * 256 * 16];
  const int tid = threadIdx.x;
  // AS3 offset: low 32 bits of the flat shared-aperture address are the LDS
  // byte address the async ops use.
  const unsigned lds0 = (unsigned)(size_t)(void*)&smem[0] + (unsigned)tid * 16u;
  const unsigned long long src_base = (unsigned long long)(uintptr_t)WtB;
  const unsigned long long dst_base = (unsigned long long)(uintptr_t)out;
  const int rstride = (int)gridDim.x * TE_RPB;

  // Issue one group of async loads into buffer `buf` (wave-uniform guards).
  auto issue_loads = [&](int rb, int buf) {
#pragma unroll
    for (int j = 0; j < TE_RPB; ++j) {
      const int r = rb + j;
      if (r < nrows) {
        const int t = ts[r];  // uniform across the wave -> broadcast load
        const unsigned srcoff = (unsigned)t * 4096u + (unsigned)tid * 16u;
        const unsigned ldsa = lds0 + (unsigned)(buf * TE_RPB + j) * 4096u;
        asm volatile("global_load_async_to_lds_b128 %0, %1, %2"
                     :
                     : "v"(ldsa), "v"(srcoff), "s"(src_base)
                     : "memory");
      }
    }
  };
  // Issue one group of async stores from buffer `buf`; non-temporal: the
  // output is write-once, keep it from evicting the L2-resident WtB table.
  auto issue_stores = [&](int rb, int buf) {
#pragma unroll
    for (int j = 0; j < TE_RPB; ++j) {
      const int r = rb + j;
      if (r < nrows) {
        const unsigned dstoff = (unsigned)r * 4096u + (unsigned)tid * 16u;
        const unsigned ldsa = lds0 + (unsigned)(buf * TE_RPB + j) * 4096u;
        asm volatile("global_store_async_from_lds_b128 %0, %1, %2 th:TH_STORE_NT"
                     :
                     : "v"(dstoff), "v"(ldsa), "s"(dst_base)
                     : "memory");
      }
    }
  };

  int rb = (int)blockIdx.x * TE_RPB;
  if (rb >= nrows) return;

  int buf = 0;
  issue_loads(rb, buf);  // prologue
  while (rb < nrows) {
    // Loads of group `rb` are in LDS; stores of group `rb - rstride` (last
    // readers of buffer buf^1) have completed -> buf^1 is reusable.
    asm volatile("s_wait_asynccnt 0x0" ::: "memory");
    issue_stores(rb, buf);
    const int rn = rb + rstride;
    if (rn < nrows) issue_loads(rn, buf ^ 1);  // overlaps the store drain
    rb = rn;
    buf ^= 1;
  }
  // Tail stores drained by s_endpgm's implicit wait-idle.
}

// ---- Fallback (workspace too small / unexpected shape): direct strided
// gather from W. Reads uncoalesced but L2-resident; stores coalesced.
__global__ void te_gather_direct(const float* __restrict__ W,
                                 const float* __restrict__ b,
                                 const int* __restrict__ ts,
                                 float* __restrict__ out,
                                 int maxt, int embed, int nrows) {
  for (int r = blockIdx.x; r < nrows; r += gridDim.x) {
    const int t = ts[r];
    for (int e = threadIdx.x; e < embed; e += blockDim.x)
      out[(size_t)r * embed + e] = W[(size_t)e * maxt + t] + b[e];
  }
}

extern "C" void kernel_launch(void* const* d_in, const int* in_sizes, int n_in,
                              void* d_out, int out_size, void* d_ws, size_t ws_size,
                              hipStream_t stream) {
  const int*   ts = (const int*)d_in[0];     // timestep [B]
  const float* W  = (const float*)d_in[1];   // [EMBED, MAXT]
  const float* b  = (const float*)d_in[2];   // [EMBED]
  float* out = (float*)d_out;

  const int nrows = in_sizes[0];             // 262144
  const int embed = in_sizes[2];             // 1024
  const int maxt  = in_sizes[1] / embed;     // 1000
  const size_t ws_need = (size_t)maxt * (size_t)embed * sizeof(float);  // 4 MB

  if (embed == 1024 && ws_size >= ws_need && d_ws != nullptr) {
    float* WtB = (float*)d_ws;
    // 1) transpose + bias fold (trivial cost)
    dim3 tb(32, 8, 1);
    dim3 tg((maxt + 31) / 32, (embed + 31) / 32, 1);
    te_transpose_bias<<<tg, tb, 0, stream>>>(W, b, WtB, maxt, embed);
    // 2) bandwidth-bound async-DMA row gather, double-buffered pipeline
    te_gather_rows_async<<<2048, 256, 0, stream>>>(WtB, ts, out, nrows);
  } else {
    te_gather_direct<<<2048, 256, 0, stream>>>(W, b, ts, out, maxt, embed, nrows);
  }
  (void)n_in; (void)out_size;
}